// MaskedQueryAndGroup_55997783605460
// MI455X (gfx1250) — compile-verified
//
#include <hip/hip_runtime.h>

namespace {
constexpr int B_ = 4, M_ = 2048, N_ = 8192, C_ = 64, K_ = 32;
constexpr int WAVES = 4;
constexpr float R2 = 0.04f;
}

typedef __attribute__((ext_vector_type(2))) float v2f;
typedef __attribute__((ext_vector_type(8))) float v8f;

__global__ __launch_bounds__(WAVES * 32) void mqag_kernel(
    const float* __restrict__ qxyz,
    const float* __restrict__ sxyz,
    const unsigned char* __restrict__ qmask,
    const unsigned char* __restrict__ smask,
    const float* __restrict__ feat,
    float* __restrict__ out)
{
  __shared__ float sc[WAVES][32][17];      // score tile, padded stride 17 (conflict-free)
  __shared__ float lds_d[WAVES][K_][32];   // sorted dists, [k][lane]
  __shared__ int   lds_i[WAVES][K_][32];   // sorted indices

  const int lane = threadIdx.x & 31;
  const int wave = threadIdx.x >> 5;
  const int task = blockIdx.x * WAVES + wave;   // 256 tasks: 64 query-tiles x 4 batches
  const int b    = task >> 6;
  const int m0   = (task & 63) * 32;
  const int ar   = lane & 15;
  const bool hi  = lane >= 16;

  // This lane's query point (row owner for selection + output)
  const int m = m0 + lane;
  const float qx = qxyz[((size_t)b * M_ + m) * 3 + 0];
  const float qy = qxyz[((size_t)b * M_ + m) * 3 + 1];
  const float qz = qxyz[((size_t)b * M_ + m) * 3 + 2];
  const float thresh = R2 - (qx*qx + qy*qy + qz*qz);  // score < thresh  <=>  d2 < r2

  // A operands: 16x4 f32 per ISA layout. lanes 0-15: (K0,K1)=(-2x,-2y); lanes 16-31: (K2,K3)=(-2z, 1)
  v2f a0, a1;
  {
    const float* q = &qxyz[((size_t)b * M_ + (m0 + ar)) * 3];
    float x = q[0], y = q[1], z = q[2];
    a0.x = hi ? (-2.0f * z) : (-2.0f * x);
    a0.y = hi ? 1.0f        : (-2.0f * y);
    const float* q2 = &qxyz[((size_t)b * M_ + (m0 + 16 + ar)) * 3];
    x = q2[0]; y = q2[1]; z = q2[2];
    a1.x = hi ? (-2.0f * z) : (-2.0f * x);
    a1.y = hi ? 1.0f        : (-2.0f * y);
  }

  int cnt = 0;
  float worst = __builtin_inff();

  for (int n0 = 0; n0 < N_; n0 += 16) {
    // B operand: 4x16 f32. lanes 0-15: (K0,K1)=(sx,sy); lanes 16-31: (K2,K3)=(sz,|s|^2)
    const int col = n0 + ar;
    const float* sp = &sxyz[((size_t)b * N_ + col) * 3];
    const float x = sp[0], y = sp[1], z = sp[2];
    const float n2 = smask[(size_t)b * N_ + col] ? (x*x + y*y + z*z) : __builtin_inff();
    v2f bb;
    bb.x = hi ? z  : x;
    bb.y = hi ? n2 : y;

    v8f cz = {0.f, 0.f, 0.f, 0.f, 0.f, 0.f, 0.f, 0.f};
    // score tile = -2 q.s + |s|^2   (16x16 per WMMA, two row-tiles)
    v8f c0 = __builtin_amdgcn_wmma_f32_16x16x4_f32(false, a0, false, bb, (short)0, cz, false, false);
    v8f c1 = __builtin_amdgcn_wmma_f32_16x16x4_f32(false, a1, false, bb, (short)0, cz, false, false);

    // D layout: VGPR v -> row v (+8 for lanes 16-31), col = lane&15
    const int r0 = hi ? 8 : 0;
#pragma unroll
    for (int v = 0; v < 8; ++v) {
      sc[wave][r0 + v][ar]      = c0[v];
      sc[wave][16 + r0 + v][ar] = c1[v];
    }
    // wave-local LDS is in-order; one split-counter wait makes writes visible cross-lane
    asm volatile("s_wait_dscnt 0" ::: "memory");

    // Each lane scans the 16 scores of its own query row; rare insertions into sorted list
    for (int cc = 0; cc < 16; ++cc) {
      const float sv = sc[wave][lane][cc];
      if (sv < thresh && sv < worst) {
        int pos = (cnt < K_) ? cnt : (K_ - 1);
        while (pos > 0) {
          float dp = lds_d[wave][pos - 1][lane];
          if (dp <= sv) break;                    // stable: ties keep earlier (lower) index
          lds_d[wave][pos][lane] = dp;
          lds_i[wave][pos][lane] = lds_i[wave][pos - 1][lane];
          --pos;
        }
        lds_d[wave][pos][lane] = sv;
        lds_i[wave][pos][lane] = n0 + cc;
        if (cnt < K_) ++cnt;
        if (cnt == K_) worst = lds_d[wave][K_ - 1][lane];
      }
    }
    asm volatile("s_wait_dscnt 0" ::: "memory");
  }

  // Neighbor indices -> registers (fully unrolled, constant indexing)
  int jreg[K_];
  const int j0 = (cnt > 0) ? lds_i[wave][0][lane] : 0;
#pragma unroll
  for (int k = 0; k < K_; ++k)
    jreg[k] = (k < cnt) ? lds_i[wave][k][lane] : j0;

  float* out1 = out;                                    // [B][3+C][M][K]
  float* out2 = out + (size_t)B_ * (3 + C_) * M_ * K_;  // mask [B][M][K]

  // xyz channels: grouped_xyz - query
  const float* sb = sxyz + (size_t)b * N_ * 3;
  {
    size_t base = (((size_t)b * (3 + C_) + 0) * M_ + m) * K_;
#pragma unroll
    for (int k = 0; k < K_; ++k) out1[base + k] = sb[jreg[k] * 3 + 0] - qx;
    base += (size_t)M_ * K_;
#pragma unroll
    for (int k = 0; k < K_; ++k) out1[base + k] = sb[jreg[k] * 3 + 1] - qy;
    base += (size_t)M_ * K_;
#pragma unroll
    for (int k = 0; k < K_; ++k) out1[base + k] = sb[jreg[k] * 3 + 2] - qz;
  }

  // feature channels: L2-resident gather, 128B contiguous stores per lane per channel
  for (int c = 0; c < C_; ++c) {
    const float* f = feat + ((size_t)b * C_ + c) * N_;
    size_t base = (((size_t)b * (3 + C_) + 3 + c) * M_ + m) * K_;
#pragma unroll
    for (int k = 0; k < K_; k += 4) {
      float4 v;
      v.x = f[jreg[k + 0]];
      v.y = f[jreg[k + 1]];
      v.z = f[jreg[k + 2]];
      v.w = f[jreg[k + 3]];
      *reinterpret_cast<float4*>(&out1[base + k]) = v;
    }
  }

  // validity mask (float 0/1)
  const bool qv = qmask[(size_t)b * M_ + m] != 0;
  const size_t mb = ((size_t)b * M_ + m) * K_;
#pragma unroll
  for (int k = 0; k < K_; ++k)
    out2[mb + k] = (qv && (k < cnt)) ? 1.0f : 0.0f;
}

extern "C" void kernel_launch(void* const* d_in, const int* in_sizes, int n_in,
                              void* d_out, int out_size, void* d_ws, size_t ws_size,
                              hipStream_t stream) {
  (void)in_sizes; (void)n_in; (void)out_size; (void)d_ws; (void)ws_size;
  const float* qxyz = (const float*)d_in[0];
  const float* sxyz = (const float*)d_in[1];
  const unsigned char* qmask = (const unsigned char*)d_in[2];
  const unsigned char* smask = (const unsigned char*)d_in[3];
  const float* feat = (const float*)d_in[4];
  float* out = (float*)d_out;

  const int tasks = B_ * (M_ / 32);   // 256 wave-tasks
  dim3 grid(tasks / WAVES);           // 64 blocks
  dim3 block(WAVES * 32);             // 128 threads = 4 waves
  hipLaunchKernelGGL(mqag_kernel, grid, block, 0, stream,
                     qxyz, sxyz, qmask, smask, feat, out);
}